// RetrievalMemory_30021821399691
// MI455X (gfx1250) — compile-verified
//
#include <hip/hip_runtime.h>
#include <stdint.h>

// Shapes are fixed by the reference.
#define Bq 4
#define Tq 4096
#define Dq 768
#define Sq 1024
#define S2 2048      // 2*S slots
#define TOPK 8
#define RWEIGHT 0.1f
#define QSCALE 0.03608439182435161f   // 1/sqrt(768)

typedef __attribute__((ext_vector_type(2))) float v2f;
typedef __attribute__((ext_vector_type(8))) float v8f;

// ---------------------------------------------------------------------------
// Kernel 0: 768x768 transpose (LDS-tiled), Wt[n][k] = W[k][n].
// Makes the (k,k+1) B-fragment pair contiguous so the GEMM LDS reads are
// single aligned ds_load_b64 with no repacking.
// ---------------------------------------------------------------------------
__global__ void __launch_bounds__(256)
transpose768(const float* __restrict__ in, float* __restrict__ outp) {
  __shared__ float tile[32][33];
  const int bx = blockIdx.x * 32;        // n block
  const int by = blockIdx.y * 32;        // k block
  const int tx = threadIdx.x & 31;
  const int ty = threadIdx.x >> 5;       // 32x8 threads
#pragma unroll
  for (int r = 0; r < 32; r += 8)
    tile[ty + r][tx] = in[(size_t)(by + ty + r) * Dq + bx + tx];
  __syncthreads();
#pragma unroll
  for (int r = 0; r < 32; r += 8)
    outp[(size_t)(bx + ty + r) * Dq + by + tx] = tile[tx][ty + r];
}

// ---------------------------------------------------------------------------
// Kernel 1: build slots[B,2S,D] = concat(mean-pool(x, window=4), persistent)
// ---------------------------------------------------------------------------
__global__ void __launch_bounds__(256)
build_slots_kernel(const float* __restrict__ x, const float* __restrict__ pers,
                   float* __restrict__ slots) {
  const int r = blockIdx.x;          // 0 .. B*2S-1
  const int b = r >> 11;             // /2048
  const int s = r & 2047;
  float* dst = slots + (size_t)r * Dq;
  if (s < Sq) {
    const float* xp = x + ((size_t)(b * Tq + s * 4)) * Dq;
    for (int d = threadIdx.x; d < Dq; d += 256)
      dst[d] = 0.25f * (xp[d] + xp[d + Dq] + xp[d + 2 * Dq] + xp[d + 3 * Dq]);
  } else {
    const float* pp = pers + (size_t)(s - Sq) * Dq;
    for (int d = threadIdx.x; d < Dq; d += 256)
      dst[d] = pp[d];
  }
}

// ---------------------------------------------------------------------------
// Kernel 2: fp32 WMMA GEMM  C[M,N] = alpha * (A[M,K] @ W[K,N] + bias[N]),
// with W supplied TRANSPOSED (Wt[N][K], row stride Kd).
// Block = 8 waves -> 128x64 C tile; each wave owns 16x64 (4 accumulator sets,
// A-fragment reused in registers across the 4 N-subtiles). The shared Wt chunk
// [64 n x 32 k] is staged into LDS with global_load_async_to_lds_b128
// (ASYNCcnt); B fragments then read as single aligned ds_load_b64.
// M % 128 == 0, N % 64 == 0, K % 32 == 0 (true for all uses here).
// ---------------------------------------------------------------------------
__global__ void __launch_bounds__(256)
gemm_wmma_f32(const float* __restrict__ A, const float* __restrict__ Wt,
              const float* __restrict__ bias, float* __restrict__ C,
              int M, int N, int Kd, float alpha) {
  __shared__ float wbuf[64 * 32];               // 8 KB Wt chunk, n-major
  const int tid  = threadIdx.x;
  const int lane = tid & 31;
  const int wave = tid >> 5;
  const int m0   = (blockIdx.x * 8 + wave) * 16;
  const int n0   = blockIdx.y * 64;
  const int half = lane >> 4;                   // 0: lanes 0-15, 1: lanes 16-31
  const int l15  = lane & 15;
  const size_t arow = (size_t)(m0 + l15) * Kd;

  // LDS byte address of wbuf: low 32 bits of the flat address are the LDS
  // offset (flat->LDS aperture mapping uses addr[31:0]).
  const uint32_t lds_base = (uint32_t)(uintptr_t)wbuf;
  const int s0 = tid, s1 = tid + 256;           // 512 x 16B segments per chunk
  const uint32_t l0 = lds_base + (uint32_t)s0 * 16u;
  const uint32_t l1 = lds_base + (uint32_t)s1 * 16u;
  // segment s covers Wt[n0 + (s>>3)][kc + (s&7)*4 .. +3]
  const uint32_t gr0 = (uint32_t)(((n0 + (s0 >> 3)) * Kd + (s0 & 7) * 4) * 4);
  const uint32_t gr1 = (uint32_t)(((n0 + (s1 >> 3)) * Kd + (s1 & 7) * 4) * 4);

  v8f acc0 = {}, acc1 = {}, acc2 = {}, acc3 = {};

  for (int kc = 0; kc < Kd; kc += 32) {
    __syncthreads();                            // previous chunk fully consumed
    // ---- stage Wt[n0..n0+63][kc..kc+31] -> wbuf (2 async b128 per thread)
    {
      const uint32_t g0 = gr0 + (uint32_t)(kc * 4);
      const uint32_t g1 = gr1 + (uint32_t)(kc * 4);
      asm volatile("global_load_async_to_lds_b128 %0, %1, %2\n\t"
                   "global_load_async_to_lds_b128 %3, %4, %2\n\t"
                   "s_wait_asynccnt 0x0"
                   :: "v"(l0), "v"(g0), "s"(Wt), "v"(l1), "v"(g1)
                   : "memory");
    }
    __syncthreads();                            // chunk visible block-wide
    // ---- 8 k-steps of 16x16x4 WMMA; A frag reused across 4 N-subtiles
#pragma unroll
    for (int kk = 0; kk < 32; kk += 4) {
      const int ka = kk + half * 2;
      v2f a;
      a.x = A[arow + kc + ka];
      a.y = A[arow + kc + ka + 1];
      v2f b0, b1, b2, b3;
      const float* p0 = &wbuf[( 0 + l15) * 32 + ka];
      const float* p1 = &wbuf[(16 + l15) * 32 + ka];
      const float* p2 = &wbuf[(32 + l15) * 32 + ka];
      const float* p3 = &wbuf[(48 + l15) * 32 + ka];
      b0.x = p0[0]; b0.y = p0[1];
      b1.x = p1[0]; b1.y = p1[1];
      b2.x = p2[0]; b2.y = p2[1];
      b3.x = p3[0]; b3.y = p3[1];
      acc0 = __builtin_amdgcn_wmma_f32_16x16x4_f32(false, a, false, b0, (short)0, acc0, false, false);
      acc1 = __builtin_amdgcn_wmma_f32_16x16x4_f32(false, a, false, b1, (short)0, acc1, false, false);
      acc2 = __builtin_amdgcn_wmma_f32_16x16x4_f32(false, a, false, b2, (short)0, acc2, false, false);
      acc3 = __builtin_amdgcn_wmma_f32_16x16x4_f32(false, a, false, b3, (short)0, acc3, false, false);
    }
  }

  // ---- epilogue: 4 subtiles of 16x16
#pragma unroll
  for (int j = 0; j < 4; ++j) {
    const v8f acc = (j == 0) ? acc0 : (j == 1) ? acc1 : (j == 2) ? acc2 : acc3;
    const int col = n0 + j * 16 + l15;
    const float bv = bias[col];
#pragma unroll
    for (int r = 0; r < 8; ++r)
      C[(size_t)(m0 + r + half * 8) * N + col] = alpha * (acc[r] + bv);
  }
}

// ---------------------------------------------------------------------------
// Kernel 3: fused scores (WMMA) + exact top-8 + softmax + gather.
// One block = 16 query tokens of one batch. Scores stripe [16,2048] lives in
// dynamic LDS (128 KB). 8 waves each own a 256-slot column chunk.
// ---------------------------------------------------------------------------
__global__ void __launch_bounds__(256)
scores_topk_gather(const float* __restrict__ Q,   // [B*T, D], pre-scaled by 1/sqrt(D)
                   const float* __restrict__ Km,  // [B*2S, D]
                   const float* __restrict__ Vm,  // [B*2S, D]
                   float* __restrict__ R) {       // [B*T, D] retrieved
  extern __shared__ float smem[];                 // [16*2048] scores + tails
  float* sw = smem + 16 * S2;                     // [16][8] weights
  int*   si = (int*)(sw + 16 * TOPK);             // [16][8] indices

  const int b    = blockIdx.x >> 8;               // T/16 = 256 blocks per batch
  const int t0   = (blockIdx.x & 255) * 16;
  const int lane = threadIdx.x & 31;
  const int wave = threadIdx.x >> 5;
  const int half = lane >> 4;
  const int l15  = lane & 15;
  const size_t qrow = (size_t)(b * Tq + t0 + l15) * Dq;
  const size_t kbase = (size_t)b * S2;

  // ---- phase 1: scores[16][2048] = Qblk @ K^T  (16 tiles of 16x16 per wave)
  for (int j = 0; j < 16; ++j) {
    const int n0 = wave * 256 + j * 16;
    const size_t krow = (kbase + n0 + l15) * Dq;
    v8f acc = {};
#pragma unroll 4
    for (int k = 0; k < Dq; k += 4) {
      const int ka = k + half * 2;
      v2f a, bfr;
      a.x = Q[qrow + ka];
      a.y = Q[qrow + ka + 1];
      bfr.x = Km[krow + ka];        // B[kk][n] = K[n0+n][kk] -> contiguous pair
      bfr.y = Km[krow + ka + 1];
      acc = __builtin_amdgcn_wmma_f32_16x16x4_f32(false, a, false, bfr,
                                                  (short)0, acc, false, false);
    }
#pragma unroll
    for (int r = 0; r < 8; ++r)
      smem[(r + half * 8) * S2 + n0 + l15] = acc[r];
  }
  __syncthreads();

  // ---- phase 2: exact top-8 per row (wave per row; each wave does 2 rows)
  for (int rr = 0; rr < 2; ++rr) {
    const int row = wave + rr * 8;
    float lv[TOPK];
    int   li[TOPK];
#pragma unroll
    for (int i = 0; i < TOPK; ++i) { lv[i] = -__builtin_inff(); li[i] = 0x7fffffff; }
    for (int s = lane; s < S2; s += 32) {
      const float v = smem[row * S2 + s];
      if (v > lv[TOPK - 1]) {                     // insertion into sorted list
        int p = TOPK - 1;
        while (p > 0 && lv[p - 1] < v) { lv[p] = lv[p - 1]; li[p] = li[p - 1]; --p; }
        lv[p] = v; li[p] = s;
      }
    }
    // 8 rounds of wave arg-max over each lane's current head (tie: lower index)
    int ptr = 0;
    float topv[TOPK]; int topi[TOPK];
    for (int kk = 0; kk < TOPK; ++kk) {
      float cv = (ptr < TOPK) ? lv[ptr] : -__builtin_inff();
      int   ci = (ptr < TOPK) ? li[ptr] : 0x7fffffff;
      int   cl = lane;
#pragma unroll
      for (int off = 16; off > 0; off >>= 1) {
        const float ov = __shfl_xor(cv, off);
        const int   oi = __shfl_xor(ci, off);
        const int   ol = __shfl_xor(cl, off);
        if (ov > cv || (ov == cv && oi < ci)) { cv = ov; ci = oi; cl = ol; }
      }
      topv[kk] = cv; topi[kk] = ci;
      if (lane == cl) ++ptr;                      // winner pops its head
    }
    // softmax over the 8 (sorted descending -> topv[0] is max)
    float sum = 0.f, wv[TOPK];
#pragma unroll
    for (int kk = 0; kk < TOPK; ++kk) { wv[kk] = __expf(topv[kk] - topv[0]); sum += wv[kk]; }
    const float inv = 1.0f / sum;
    if (lane < TOPK) { sw[row * TOPK + lane] = wv[lane] * inv; si[row * TOPK + lane] = topi[lane]; }
  }
  __syncthreads();

  // ---- phase 3: retrieved[16][768] = sum_k w * V[idx]
  for (int e = threadIdx.x; e < 16 * Dq; e += 256) {
    const int row = e / Dq, col = e % Dq;
    float acc = 0.f;
#pragma unroll
    for (int kk = 0; kk < TOPK; ++kk)
      acc += sw[row * TOPK + kk] * Vm[(kbase + si[row * TOPK + kk]) * Dq + col];
    R[(size_t)(b * Tq + t0 + row) * Dq + col] = acc;
  }
}

// ---------------------------------------------------------------------------
extern "C" void kernel_launch(void* const* d_in, const int* in_sizes, int n_in,
                              void* d_out, int out_size, void* d_ws, size_t ws_size,
                              hipStream_t stream) {
  const float* x    = (const float*)d_in[0];
  const float* Wqp  = (const float*)d_in[1];
  const float* bq   = (const float*)d_in[2];
  const float* Wkp  = (const float*)d_in[3];
  const float* bk   = (const float*)d_in[4];
  const float* Wvp  = (const float*)d_in[5];
  const float* bv   = (const float*)d_in[6];
  const float* Wpp  = (const float*)d_in[7];
  const float* bp   = (const float*)d_in[8];
  const float* pers = (const float*)d_in[9];
  float* out = (float*)d_out;

  // workspace layout (floats): slots | q | k | v | retrieved | 4x Wt (~177 MiB)
  float* ws    = (float*)d_ws;
  float* slots = ws;                               // B*2S*D = 6291456
  float* qbuf  = slots + (size_t)Bq * S2 * Dq;     // B*T*D  = 12582912
  float* kbuf  = qbuf  + (size_t)Bq * Tq * Dq;     // 6291456
  float* vbuf  = kbuf  + (size_t)Bq * S2 * Dq;     // 6291456
  float* retr  = vbuf  + (size_t)Bq * S2 * Dq;     // 12582912
  float* wtq   = retr  + (size_t)Bq * Tq * Dq;     // 589824 each
  float* wtk   = wtq   + (size_t)Dq * Dq;
  float* wtv   = wtk   + (size_t)Dq * Dq;
  float* wtp   = wtv   + (size_t)Dq * Dq;

  // 0) pre-transpose the four weight matrices (tiny, one-time per launch)
  const dim3 tgrid(Dq / 32, Dq / 32);
  transpose768<<<tgrid, 256, 0, stream>>>(Wqp, wtq);
  transpose768<<<tgrid, 256, 0, stream>>>(Wkp, wtk);
  transpose768<<<tgrid, 256, 0, stream>>>(Wvp, wtv);
  transpose768<<<tgrid, 256, 0, stream>>>(Wpp, wtp);

  // 1) slots = [pool(x) ; persistent]
  build_slots_kernel<<<Bq * S2, 256, 0, stream>>>(x, pers, slots);

  // 2) projections (fold 1/sqrt(D) into q so scores kernel is pure q@k^T)
  gemm_wmma_f32<<<dim3((Bq * Tq) / 128, Dq / 64), 256, 0, stream>>>(
      x, wtq, bq, qbuf, Bq * Tq, Dq, Dq, QSCALE);
  gemm_wmma_f32<<<dim3((Bq * S2) / 128, Dq / 64), 256, 0, stream>>>(
      slots, wtk, bk, kbuf, Bq * S2, Dq, Dq, 1.0f);
  gemm_wmma_f32<<<dim3((Bq * S2) / 128, Dq / 64), 256, 0, stream>>>(
      slots, wtv, bv, vbuf, Bq * S2, Dq, Dq, 1.0f);

  // 3) fused scores + top-8 + softmax + gather (128 KB LDS stripe + tails)
  const size_t shmem = (size_t)(16 * S2 + 16 * TOPK) * sizeof(float)
                     + (size_t)(16 * TOPK) * sizeof(int);
  scores_topk_gather<<<Bq * (Tq / 16), 256, shmem, stream>>>(qbuf, kbuf, vbuf, retr);

  // 4) out = RW * (retrieved @ Wp + bp)
  gemm_wmma_f32<<<dim3((Bq * Tq) / 128, Dq / 64), 256, 0, stream>>>(
      retr, wtp, bp, out, Bq * Tq, Dq, Dq, RWEIGHT);
}